// DropStripes_56959856279685
// MI455X (gfx1250) — compile-verified
//
#include <hip/hip_runtime.h>

// DropStripes: x (64, 2048, 512) fp32, zero two random column stripes per batch.
// Memory-bound: 512 MB traffic -> ~22 us floor at 23.3 TB/s. Strategy:
// b128 non-temporal streaming, wave-uniform (scalar) stripe parameters.

typedef float v4f __attribute__((ext_vector_type(4)));

namespace {
constexpr int kB = 64;            // batches
constexpr int kH = 2048;          // rows
constexpr int kW = 512;           // width (stripe axis)
constexpr int kDropWidth = 64;
constexpr int kVec4PerRow = kW / 4;                                // 128
constexpr long long kVec4PerBatch = (long long)kH * kVec4PerRow;   // 262144
constexpr int kThreads = 256;
constexpr int kVec4PerThread = 4;                                  // 64 B / thread
constexpr int kVec4PerBlock = kThreads * kVec4PerThread;           // 1024
constexpr int kBlocksPerBatch = (int)(kVec4PerBatch / kVec4PerBlock); // 256
} // namespace

__global__ __launch_bounds__(kThreads)
void drop_stripes_kernel(const v4f* __restrict__ x,
                         const float* __restrict__ u_dist,
                         const float* __restrict__ u_bgn,
                         v4f* __restrict__ out) {
    // Each block covers 1024 consecutive float4s of exactly one batch, so
    // `b` (and the stripe parameters) are wave-uniform -> SALU/SMEM path.
    const int b = (int)(blockIdx.x / kBlocksPerBatch);

    const int d0 = (int)floorf(u_dist[2 * b + 0] * (float)kDropWidth);
    const int d1 = (int)floorf(u_dist[2 * b + 1] * (float)kDropWidth);
    const int bg0 = (int)floorf(u_bgn[2 * b + 0] * (float)(kW - d0));
    const int bg1 = (int)floorf(u_bgn[2 * b + 1] * (float)(kW - d1));
    const int e0 = bg0 + d0;
    const int e1 = bg1 + d1;

    const long long base =
        (long long)blockIdx.x * kVec4PerBlock + (long long)threadIdx.x;

#pragma unroll
    for (int j = 0; j < kVec4PerThread; ++j) {
        const long long i4 = base + (long long)j * kThreads;
        const int col = ((int)(i4 & (long long)(kVec4PerRow - 1))) << 2;

        // Streaming (non-temporal) 128-bit load: read-once data, don't
        // pollute L2 (input is 256 MB > 192 MB L2).
        v4f v = __builtin_nontemporal_load(&x[i4]);
        v4f r;
#pragma unroll
        for (int k = 0; k < 4; ++k) {
            const int c = col + k;
            const bool drop = (c >= bg0 && c < e0) || (c >= bg1 && c < e1);
            r[k] = drop ? 0.0f : v[k];
        }
        // Write-once output: non-temporal 128-bit store.
        __builtin_nontemporal_store(r, &out[i4]);
    }
}

extern "C" void kernel_launch(void* const* d_in, const int* in_sizes, int n_in,
                              void* d_out, int out_size, void* d_ws, size_t ws_size,
                              hipStream_t stream) {
    (void)in_sizes; (void)n_in; (void)out_size; (void)d_ws; (void)ws_size;

    const v4f*  x      = (const v4f*)d_in[0];
    const float* u_dist = (const float*)d_in[1];
    const float* u_bgn  = (const float*)d_in[2];
    v4f* out = (v4f*)d_out;

    const long long totalVec4 = (long long)kB * kVec4PerBatch;  // 16,777,216
    const int grid = (int)(totalVec4 / kVec4PerBlock);          // 16,384 blocks

    drop_stripes_kernel<<<dim3(grid), dim3(kThreads), 0, stream>>>(
        x, u_dist, u_bgn, out);
}